// Generator_Harm_2843268350670
// MI455X (gfx1250) — compile-verified
//
#include <hip/hip_runtime.h>
#include <hip/hip_bf16.h>

typedef float v2f __attribute__((ext_vector_type(2)));
typedef float v8f __attribute__((ext_vector_type(8)));

#define Hc    192
#define Tc    800
#define Bc    4
#define NHc   64
#define HOPc  256
#define TSAMP (Tc*HOPc)      // 204800
#define TT    32             // time-tile per block (2 WMMA N-tiles)

#if __has_builtin(__builtin_amdgcn_global_load_async_to_lds_b32) && \
    __has_builtin(__builtin_amdgcn_s_wait_asynccnt)
#define HAVE_ASYNC_LDS 1
typedef __attribute__((address_space(1))) int gint;   // global AS pointer pointee
typedef __attribute__((address_space(3))) int lint;   // LDS AS pointer pointee
#else
#define HAVE_ASYNC_LDS 0
#endif

// ---------------------------------------------------------------------------
// Conv1d layer as WMMA GEMM:  out[co,t] = sum_{ci,k} W[co,ci,k] * in[ci,t+k-P]
// GEMM: M=192 (co), N=32 (t tile), Kdim=192*KW, via V_WMMA_F32_16X16X4_F32.
// Each of 12 waves owns one 16-row co tile and two 16x16 accumulators (the
// A fragment is shared by both N-tiles -> half the weight traffic per FLOP).
// MODE: 0 = ReLU (pre)
//       1 = LayerNorm(channels)+ReLU           (net layer 0)
//       2 = (residual + LayerNorm+ReLU) * 0.5  (net layers 1..7)
//       3 = *mask (proj, KW=1)
// ---------------------------------------------------------------------------
template<int MODE, int KW>
__global__ __launch_bounds__(384) void conv_layer(
    const float* __restrict__ in,   const float* __restrict__ w,
    const float* __restrict__ bias, const float* __restrict__ g,
    const float* __restrict__ beta, const float* __restrict__ mask,
    float* __restrict__ out)
{
    constexpr int PAD = KW / 2;
    constexpr int NC  = TT + 2 * PAD;      // tile columns incl. halo
    constexpr int KD  = Hc * KW;           // GEMM K dimension

    __shared__ float Xs[Hc * NC];          // input tile (also residual source)
    __shared__ float Ys[Hc * TT];          // conv output tile
    __shared__ float mS[TT], rS[TT];       // LN stats per position

    const int b   = blockIdx.y;
    const int t0  = blockIdx.x * TT;
    const int tid = threadIdx.x;

    // stage input tile + halo into LDS (zero-padded at sequence edges),
    // using the CDNA5 async global->LDS path when available
    for (int e = tid; e < Hc * NC; e += 384) {
        int ci = e / NC, j = e - ci * NC;
        int t  = t0 + j - PAD;
#if HAVE_ASYNC_LDS
        if (t >= 0 && t < Tc)
            __builtin_amdgcn_global_load_async_to_lds_b32(
                (gint*)(in + (size_t)(b * Hc + ci) * Tc + t),
                (lint*)(Xs + e), 0, 0);
        else
            Xs[e] = 0.0f;
#else
        float v = 0.0f;
        if (t >= 0 && t < Tc) v = in[(size_t)(b * Hc + ci) * Tc + t];
        Xs[e] = v;
#endif
    }
#if HAVE_ASYNC_LDS
    __builtin_amdgcn_s_wait_asynccnt(0);
#endif
    __syncthreads();

    const int lane   = tid & 31;
    const int wave   = tid >> 5;           // 0..11 -> co tile
    const int half   = lane >> 4;          // 0/1 (K split per ISA layout)
    const int l15    = lane & 15;
    const int coBase = wave * 16;

    v8f c0 = {};                           // positions t0 + [0,16)
    v8f c1 = {};                           // positions t0 + [16,32)
    const float* wrow = w + (size_t)(coBase + l15) * KD;  // A row (contiguous K)
    __builtin_prefetch(wrow, 0, 3);        // global_prefetch_b8

    for (int kk = 0; kk < KD; kk += 4) {
        const int kOff = kk + half * 2;
        v2f a, bm0, bm1;
        // A 16x4 fragment: lanes0-15 K={kk,kk+1}, lanes16-31 K={kk+2,kk+3}
        a.x = wrow[kOff];
        a.y = wrow[kOff + 1];
        // B 4x16 fragments: Bmat[q][pos] = Xs[(q/KW)*NC + pos + (q%KW)]
        const int q0 = kOff, q1 = kOff + 1;
        const int r0 = (q0 / KW) * NC + (q0 % KW);
        const int r1 = (q1 / KW) * NC + (q1 % KW);
        bm0.x = Xs[r0 + l15];
        bm0.y = Xs[r1 + l15];
        bm1.x = Xs[r0 + 16 + l15];
        bm1.y = Xs[r1 + 16 + l15];
        c0 = __builtin_amdgcn_wmma_f32_16x16x4_f32(false, a, false, bm0,
                                                   (short)0, c0, false, false);
        c1 = __builtin_amdgcn_wmma_f32_16x16x4_f32(false, a, false, bm1,
                                                   (short)0, c1, false, false);
    }

    // C/D layout: VGPR r -> (M = r + 8*half, N = l15); add bias while spilling
#pragma unroll
    for (int r = 0; r < 8; ++r) {
        int co = coBase + half * 8 + r;
        float bi = bias[co];
        Ys[co * TT + l15]      = c0[r] + bi;
        Ys[co * TT + 16 + l15] = c1[r] + bi;
    }
    __syncthreads();

    if (MODE == 1 || MODE == 2) {
        if (tid < TT) {                    // LN stats over 192 channels
            float s = 0.0f;
            for (int ci = 0; ci < Hc; ++ci) s += Ys[ci * TT + tid];
            float m = s * (1.0f / Hc);
            float vs = 0.0f;
            for (int ci = 0; ci < Hc; ++ci) {
                float d = Ys[ci * TT + tid] - m;
                vs += d * d;
            }
            mS[tid] = m;
            rS[tid] = rsqrtf(vs * (1.0f / Hc) + 1e-5f);
        }
        __syncthreads();
    }

    for (int e = tid; e < Hc * TT; e += 384) {
        int co = e >> 5, pos = e & 31;
        float y = Ys[e];
        float o;
        if (MODE == 0) {
            o = fmaxf(y, 0.0f);
        } else if (MODE == 3) {
            o = y * mask[b * Tc + t0 + pos];
        } else {
            y = (y - mS[pos]) * rS[pos] * g[co] + beta[co];
            y = fmaxf(y, 0.0f);
            if (MODE == 2) o = (Xs[co * NC + pos + PAD] + y) * 0.5f;
            else           o = y;
        }
        out[(size_t)(b * Hc + co) * Tc + t0 + pos] = o;
    }
}

// ---------------------------------------------------------------------------
// Post conv (H->65, K=3) + ReLU + sigmoid^ln10 transform + harmonic-amp
// normalization. One block per (b,t); thread c<65 owns output channel c.
// ---------------------------------------------------------------------------
__global__ __launch_bounds__(128) void post_params(
    const float* __restrict__ hp,  const float* __restrict__ pw,
    const float* __restrict__ pb,  const float* __restrict__ f0,
    float* __restrict__ amps)
{
    __shared__ float hps[Hc * 3];
    __shared__ float parS[65];
    __shared__ float ampS[65];
    __shared__ float sumS;

    const int b = blockIdx.y, t = blockIdx.x, tid = threadIdx.x;

    for (int e = tid; e < Hc * 3; e += 128) {
        int ci = e / 3, k = e - ci * 3;
        int tt = t + k - 1;
        hps[e] = (tt >= 0 && tt < Tc) ? hp[(size_t)(b * Hc + ci) * Tc + tt] : 0.0f;
    }
    __syncthreads();

    if (tid < 65) {
        float acc = pb[tid];
        const float* wr = pw + (size_t)tid * Hc * 3;
        for (int q = 0; q < Hc * 3; ++q) acc += wr[q] * hps[q];
        acc = fmaxf(acc, 0.0f);
        float sg = 1.0f / (1.0f + expf(-acc));
        parS[tid] = 2.0f * powf(sg, 2.302585092994046f) + 1e-7f;
    }
    __syncthreads();

    const float pitch = f0[b * Tc + t];
    if (tid >= 1 && tid < 65) {
        float aa = ((pitch * (float)tid) < 11025.0f ? 1.0f : 0.0f) + 1e-4f;
        ampS[tid] = parS[tid] * aa;
    }
    __syncthreads();
    if (tid == 0) {
        float s = 0.0f;
        for (int n = 1; n < 65; ++n) s += ampS[n];
        sumS = s;
    }
    __syncthreads();
    if (tid >= 1 && tid < 65)
        amps[((size_t)(b * Tc + t)) * NHc + (tid - 1)] = ampS[tid] / sumS * parS[0];
}

// ---------------------------------------------------------------------------
// Bit-exact replication of the reference f32 cumsum: one sequential dependency
// chain of 204800 f32 adds per batch (FP add is non-associative, so this is
// the only order-faithful evaluation). float4 stores hide store traffic.
// ---------------------------------------------------------------------------
__global__ void omega_scan(const float* __restrict__ f0, float* __restrict__ om)
{
    if (threadIdx.x != 0) return;
    const int b = blockIdx.x;
    const float* fb = f0 + b * Tc;
    float4* og = (float4*)(om + (size_t)b * TSAMP);
    float acc = 0.0f;
    for (int f = 0; f < Tc; ++f) {
        float step = (6.283185307179586f * fb[f]) / 22050.0f;
        for (int k = 0; k < HOPc; k += 4) {
            float4 v;
            acc += step; v.x = acc;
            acc += step; v.y = acc;
            acc += step; v.z = acc;
            acc += step; v.w = acc;
            og[(f * HOPc + k) >> 2] = v;
        }
    }
}

// ---------------------------------------------------------------------------
// sig[b,n,t] = sin(omega[b,t] * (n+1)) * amp[b, t/HOP, n]; float4 stores.
// 210 MB output -> memory/transcendental bound; fully coalesced.
// ---------------------------------------------------------------------------
__global__ __launch_bounds__(256) void synth(
    const float* __restrict__ om, const float* __restrict__ amps,
    float* __restrict__ out)
{
    const int b  = blockIdx.z;
    const int n  = blockIdx.y + 1;                  // harmonic index 1..64
    const int i0 = (blockIdx.x * 256 + threadIdx.x) * 4;
    const int f  = i0 >> 8;                         // frame (4 samples never cross)

    const float4 o4 = *(const float4*)(om + (size_t)b * TSAMP + i0);
    const float amp = amps[((size_t)(b * Tc + f)) * NHc + (n - 1)];
    const float fn  = (float)n;

    float4 r;
    r.x = sinf(o4.x * fn) * amp;
    r.y = sinf(o4.y * fn) * amp;
    r.z = sinf(o4.z * fn) * amp;
    r.w = sinf(o4.w * fn) * amp;
    *(float4*)(out + ((size_t)(b * NHc + (n - 1))) * TSAMP + i0) = r;
}

// ---------------------------------------------------------------------------
extern "C" void kernel_launch(void* const* d_in, const int* in_sizes, int n_in,
                              void* d_out, int out_size, void* d_ws, size_t ws_size,
                              hipStream_t stream)
{
    const float* f0     = (const float*)d_in[0];
    const float* harm   = (const float*)d_in[1];
    const float* mask   = (const float*)d_in[2];
    const float* pre_w  = (const float*)d_in[3];
    const float* pre_b  = (const float*)d_in[4];
    const float* net_w  = (const float*)d_in[5];
    const float* net_b  = (const float*)d_in[6];
    const float* ln_g   = (const float*)d_in[7];
    const float* ln_b   = (const float*)d_in[8];
    const float* proj_w = (const float*)d_in[9];
    const float* proj_b = (const float*)d_in[10];
    const float* post_w = (const float*)d_in[11];
    const float* post_b = (const float*)d_in[12];

    float* ws   = (float*)d_ws;
    const size_t nAct = (size_t)Bc * Hc * Tc;       // 614400
    float* buf0 = ws;
    float* buf1 = ws + nAct;
    float* ampb = ws + 2 * nAct;                    // B*T*NH
    float* omg  = ampb + (size_t)Bc * Tc * NHc;     // B*TSAMP
    float* outp = (float*)d_out;

    const dim3 cg(Tc / TT, Bc);

    // pre conv + ReLU
    conv_layer<0, 3><<<cg, 384, 0, stream>>>(harm, pre_w, pre_b,
                                             nullptr, nullptr, nullptr, buf0);
    // net layer 0: conv + LN + ReLU
    conv_layer<1, 3><<<cg, 384, 0, stream>>>(buf0, net_w, net_b, ln_g, ln_b,
                                             nullptr, buf1);
    // net layers 1..7: h = (h + relu(ln(conv(h)))) * 0.5, ping-pong buffers
    const float* cur = buf1;
    float* nxt = buf0;
    for (int i = 1; i < 8; ++i) {
        conv_layer<2, 3><<<cg, 384, 0, stream>>>(
            cur, net_w + (size_t)i * Hc * Hc * 3, net_b + i * Hc,
            ln_g + i * Hc, ln_b + i * Hc, nullptr, nxt);
        float* tmp = (float*)cur; cur = nxt; nxt = tmp;
    }
    // after 7 swaps: cur == buf0, nxt == buf1

    // proj (K=1) + mask
    conv_layer<3, 1><<<cg, 384, 0, stream>>>(cur, proj_w, proj_b,
                                             nullptr, nullptr, mask, nxt);
    // post conv + param transform + amp normalization
    post_params<<<dim3(Tc, Bc), 128, 0, stream>>>(nxt, post_w, post_b, f0, ampb);
    // exact sequential phase accumulation
    omega_scan<<<Bc, 32, 0, stream>>>(f0, omg);
    // harmonic synthesis -> d_out (B, NH, TSAMP)
    synth<<<dim3(TSAMP / 1024, NHc, Bc), 256, 0, stream>>>(omg, ampb, outp);

    (void)in_sizes; (void)n_in; (void)out_size; (void)ws_size;
}